// RadialSpectrumLoss_28595892257477
// MI455X (gfx1250) — compile-verified
//
#include <hip/hip_runtime.h>
#include <hip/hip_bf16.h>
#include <math.h>

// ---------------------------------------------------------------------------
// RadialSpectrumLoss on gfx1250 (MI455X)
//   density[b] = gy^T @ gx  -> f32 WMMA (V_WMMA_F32_16X16X4_F32)
//   FFT2 via shared-memory radix-2, |F|^2, fftshift, radial bin means, L1 diff
// ---------------------------------------------------------------------------

typedef float v2f __attribute__((ext_vector_type(2)));
typedef float v8f __attribute__((ext_vector_type(8)));

#define IMG 256        // H == W
#define NPTS 1024      // N
#define NIMG 8         // 4 pred batches + 4 gt batches
#define NBINS 64

// ---------------------------------------------------------------------------
// Kernel 1: Gaussian factor tables
//   gx[s][n][w] = exp(-(w/255 - px)^2 * 8192), gy likewise with py.
//   s in [0,8): s<4 -> pred batch s ; s>=4 -> gt batch s-4.
// ---------------------------------------------------------------------------
__global__ void rs_gauss(const float* __restrict__ pred,
                         const float* __restrict__ gt,
                         float* __restrict__ gx,
                         float* __restrict__ gy) {
    int s = blockIdx.x >> 10;          // image 0..7
    int n = blockIdx.x & 1023;         // point 0..1023
    const float* pts = (s < 4) ? pred : gt;
    int b = s & 3;
    float px = pts[(b * NPTS + n) * 2 + 0];
    float py = pts[(b * NPTS + n) * 2 + 1];
    int w = threadIdx.x;               // 0..255
    float coord = (float)w * (1.0f / 255.0f);   // linspace(0,1,256)
    float dx = coord - px;
    float dy = coord - py;
    const float inv = 8192.0f;         // 1/(2*sigma^2), sigma = 2/256
    long idx = (long)(s * NPTS + n) * IMG + w;
    gx[idx] = __expf(-dx * dx * inv);
    gy[idx] = __expf(-dy * dy * inv);
}

// ---------------------------------------------------------------------------
// Kernel 2: density[s][h][w] = sum_k gy[s][k][h] * gx[s][k][w]
//   One wave per 16x16 output tile; K=1024 in steps of 4 via f32 WMMA.
//   A (16x4, M=h K=k): lane(0..15)=M, vgpr0/1 = K{0,1} (lanes<16) / K{2,3}.
//   B (4x16, K=k N=w): lane=N with same half-wave K split.
//   C/D (16x16): vgpr r -> M=r (lanes 0..15) / M=8+r (lanes 16..31), N=lane%16.
// ---------------------------------------------------------------------------
__global__ void rs_density_wmma(const float* __restrict__ gx,
                                const float* __restrict__ gy,
                                float* __restrict__ dens) {
    int wave = threadIdx.x >> 5;                  // 8 waves / block
    int lane = threadIdx.x & 31;
    int gidx = blockIdx.x * 8 + wave;             // 0..2047 tiles
    int s    = gidx >> 8;                         // image
    int tile = gidx & 255;
    int th = (tile >> 4) << 4;                    // tile h origin
    int tw = (tile & 15) << 4;                    // tile w origin

    const float* gys = gy + (long)s * NPTS * IMG;
    const float* gxs = gx + (long)s * NPTS * IMG;

    int lh   = lane & 15;                         // M / N within tile
    int koff = (lane >> 4) << 1;                  // 0 for lanes<16, 2 else

    v8f acc = {};
    for (int k = 0; k < NPTS; k += 4) {
        v2f a, b;
        a.x = gys[(k + koff + 0) * IMG + th + lh];
        a.y = gys[(k + koff + 1) * IMG + th + lh];
        b.x = gxs[(k + koff + 0) * IMG + tw + lh];
        b.y = gxs[(k + koff + 1) * IMG + tw + lh];
        acc = __builtin_amdgcn_wmma_f32_16x16x4_f32(
            /*neg_a=*/false, a, /*neg_b=*/false, b,
            /*c_mod=*/(short)0, acc, /*reuse_a=*/false, /*reuse_b=*/false);
    }

    float* out = dens + ((long)s * IMG + th) * IMG + tw;
    int m0 = (lane >> 4) << 3;                    // 0 or 8
    #pragma unroll
    for (int r = 0; r < 8; ++r)
        out[(m0 + r) * IMG + lh] = acc[r];
}

// ---------------------------------------------------------------------------
// Radix-2 DIT butterfly stage helper (256 points in LDS, 8 stages)
// ---------------------------------------------------------------------------
__device__ __forceinline__ void fft256_stages(float2* buf, int t) {
    #pragma unroll
    for (int stage = 1; stage <= 8; ++stage) {
        int half = 1 << (stage - 1);
        int len  = 1 << stage;
        if (t < 128) {
            int k  = t & (half - 1);
            int g  = t >> (stage - 1);
            int i1 = g * len + k;
            int i2 = i1 + half;
            float ang = -3.14159265358979f * (float)k / (float)half;
            float sn, cs;
            __sincosf(ang, &sn, &cs);
            float2 u = buf[i1];
            float2 v = buf[i2];
            float2 tm;
            tm.x = v.x * cs - v.y * sn;
            tm.y = v.x * sn + v.y * cs;
            buf[i1] = make_float2(u.x + tm.x, u.y + tm.y);
            buf[i2] = make_float2(u.x - tm.x, u.y - tm.y);
        }
        __syncthreads();
    }
}

// ---------------------------------------------------------------------------
// Kernel 3: row FFTs. real density row -> complex spectrum row.
// ---------------------------------------------------------------------------
__global__ void rs_fft_rows(const float* __restrict__ dens,
                            float2* __restrict__ freq) {
    __shared__ float2 buf[IMG];
    int row = blockIdx.x;                 // s*256 + h
    int t = threadIdx.x;
    int rev = __brev((unsigned)t) >> 24;  // 8-bit bit reversal
    buf[rev] = make_float2(dens[(long)row * IMG + t], 0.0f);
    __syncthreads();
    fft256_stages(buf, t);
    freq[(long)row * IMG + t] = buf[t];
}

// ---------------------------------------------------------------------------
// Kernel 4: column FFTs + power + fftshift + radial binning.
// ---------------------------------------------------------------------------
__global__ void rs_fft_cols(const float2* __restrict__ freq,
                            const int* __restrict__ rbins,
                            float* __restrict__ binsum) {
    __shared__ float2 buf[IMG];
    __shared__ float bins[NBINS];
    int s   = blockIdx.x >> 8;
    int col = blockIdx.x & 255;
    int t = threadIdx.x;
    if (t < NBINS) bins[t] = 0.0f;
    int rev = __brev((unsigned)t) >> 24;
    buf[rev] = freq[((long)s * IMG + t) * IMG + col];
    __syncthreads();
    fft256_stages(buf, t);
    // buf[t] = F[s][t][col]; place power at fftshifted coords, accumulate bin
    float2 v = buf[t];
    float p = v.x * v.x + v.y * v.y;
    int us = (t + 128) & 255;
    int ws = (col + 128) & 255;
    atomicAdd(&bins[rbins[us * IMG + ws]], p);
    __syncthreads();
    if (t < NBINS) atomicAdd(&binsum[s * NBINS + t], bins[t]);
}

// ---------------------------------------------------------------------------
// Kernel 5: zero the bin accumulators.
// ---------------------------------------------------------------------------
__global__ void rs_zero(float* __restrict__ binsum) {
    int t = threadIdx.x + blockIdx.x * blockDim.x;
    if (t < NIMG * NBINS) binsum[t] = 0.0f;
}

// ---------------------------------------------------------------------------
// Kernel 6: per-bin mean, L1 normalize per image, mean |pred - gt|.
// ---------------------------------------------------------------------------
__global__ void rs_finalize(const int* __restrict__ rbins,
                            const float* __restrict__ binsum,
                            float* __restrict__ out) {
    __shared__ float cnt[NBINS];
    __shared__ float rs[NIMG * NBINS];
    __shared__ float tot[NIMG];
    __shared__ float part[NBINS];
    int t = threadIdx.x;
    if (t < NBINS) cnt[t] = 0.0f;
    __syncthreads();
    for (int i = t; i < IMG * IMG; i += 256)
        atomicAdd(&cnt[rbins[i]], 1.0f);
    __syncthreads();
    for (int i = t; i < NIMG * NBINS; i += 256)
        rs[i] = binsum[i] / cnt[i & (NBINS - 1)];
    __syncthreads();
    if (t < NIMG) {
        float ssum = 0.0f;
        for (int j = 0; j < NBINS; ++j) ssum += rs[t * NBINS + j];
        tot[t] = ssum + 1e-8f;
    }
    __syncthreads();
    for (int i = t; i < NIMG * NBINS; i += 256)
        rs[i] /= tot[i >> 6];
    __syncthreads();
    if (t < NBINS) {
        float d = 0.0f;
        for (int b = 0; b < 4; ++b)
            d += fabsf(rs[b * NBINS + t] - rs[(b + 4) * NBINS + t]);
        part[t] = d;
    }
    __syncthreads();
    if (t == 0) {
        float ssum = 0.0f;
        for (int j = 0; j < NBINS; ++j) ssum += part[j];
        out[0] = ssum * (1.0f / 256.0f);   // mean over 4*64 entries
    }
}

// ---------------------------------------------------------------------------
// Host-side launcher
// ---------------------------------------------------------------------------
extern "C" void kernel_launch(void* const* d_in, const int* in_sizes, int n_in,
                              void* d_out, int out_size, void* d_ws, size_t ws_size,
                              hipStream_t stream) {
    (void)in_sizes; (void)n_in; (void)out_size; (void)ws_size;

    const float* pred  = (const float*)d_in[0];   // [4,1024,2]
    const float* gt    = (const float*)d_in[1];   // [4,1024,2]
    const int*   rbins = (const int*)d_in[2];     // [256*256]
    float* out = (float*)d_out;                   // scalar

    // Workspace layout (bytes):
    //   [0, 8MB)   gx  [8][1024][256] f32  -- reused as freq (4MB) after GEMM
    //   [8, 16MB)  gy  [8][1024][256] f32
    //   [16,18MB)  dens[8][256][256]  f32
    //   [18MB, +2KB) binsum [8][64]   f32
    float*  gx     = (float*)d_ws;
    float*  gy     = gx + (long)NIMG * NPTS * IMG;
    float*  dens   = gy + (long)NIMG * NPTS * IMG;
    float*  binsum = dens + (long)NIMG * IMG * IMG;
    float2* freq   = (float2*)d_ws;               // aliases dead gx region

    rs_gauss<<<NIMG * NPTS, IMG, 0, stream>>>(pred, gt, gx, gy);
    rs_density_wmma<<<256, 256, 0, stream>>>(gx, gy, dens);
    rs_zero<<<2, 256, 0, stream>>>(binsum);
    rs_fft_rows<<<NIMG * IMG, IMG, 0, stream>>>(dens, freq);
    rs_fft_cols<<<NIMG * IMG, IMG, 0, stream>>>(freq, rbins, binsum);
    rs_finalize<<<1, 256, 0, stream>>>(rbins, binsum, out);
}